// LRGenerator_39049842655404
// MI455X (gfx1250) — compile-verified
//
#include <hip/hip_runtime.h>

// ---------------------------------------------------------------------------
// MI455X (gfx1250) implementation of the low-rank patch-token generator.
//
// Key algebraic simplification: the token projections are linear, so the
// sum-over-j (V path) / sum-over-i (H path) of projected patches equals the
// projection of the summed patches.  We therefore:
//   k1: stream y = x + pos_enc once, producing
//         vsum[bc][rp][col] = sum_j y[16j+rp][col]          (16 x 256)
//         hsum[bc][row][cp] = sum_i y[row][16i+cp]          (256 x 16)
//   k2: vcomp[bc][i][d] = Wv * vsum-patch + 16*bv   (WMMA f32 16x16x4)
//       hcomp[bc][j][d] = Wh * hsum-patch + 16*bh
//   k3: V[bc][rn] = vcomp_flat . Vw[rn] + Vb[rn]    (GEMM 256x768x1024, WMMA)
//       H[bc][rn] = hcomp_flat . Hw[rn] + Hb[rn]
//   k4: out[bc][n][m] = sum_r V[n][r] * H[m][r]     (rank-3, write-bound)
//
// Bandwidth roofline: 83 MB read (k1) + 67 MB write (k4) ~= 6.5 us @ 23.3 TB/s.
// All GEMM operands are L2-resident (<= 12 MB workspace, 192 MB L2).
// ---------------------------------------------------------------------------

typedef __attribute__((ext_vector_type(2))) float v2f;
typedef __attribute__((ext_vector_type(8))) float v8f;

#define BATCH 16
#define CHAN  16
#define NSIDE 256
#define IMG   65536            // NSIDE*NSIDE
#define NBC   256              // BATCH*CHAN
#define HD    64
#define RANKN 768              // RANK*NSIDE (flat rn = r*256+n)
#define KIN   1024             // NPATCH*HD

// workspace layout (float offsets)
#define WS_VSUM  0
#define WS_HSUM  (WS_VSUM  + NBC*4096)   // 16*256 per bc
#define WS_VCOMP (WS_HSUM  + NBC*4096)   // 256*16 per bc
#define WS_HCOMP (WS_VCOMP + NBC*1024)   // 16*64 per bc
#define WS_V     (WS_HCOMP + NBC*1024)
#define WS_H     (WS_V     + NBC*RANKN)
// total = 2*1048576 + 2*262144 + 2*196608 floats = ~12.1 MB

static __device__ inline v8f wmma_f32_k4(v2f a, v2f b, v8f c) {
    // D = A(16x4,f32) * B(4x16,f32) + C(16x16,f32)  -- V_WMMA_F32_16X16X4_F32
    return __builtin_amdgcn_wmma_f32_16x16x4_f32(false, a, false, b,
                                                 (short)0, c, false, false);
}

// ---------------------------------------------------------------------------
// k1: single-pass streaming reduction.  256 blocks (one per (b,c)), 8 waves.
// Wave w owns rows {w + 8*rr}; its row&15 values are exactly {w, w+8}, so the
// vsum accumulators are wave-private (no cross-wave combine).  hsum needs a
// 3-step wave32 shuffle tree (lanes congruent mod 4 share a cp group of 4).
// ---------------------------------------------------------------------------
__global__ __launch_bounds__(256) void k1_reduce(const float* __restrict__ x,
                                                 const float* __restrict__ pe,
                                                 float* __restrict__ vsum,
                                                 float* __restrict__ hsum) {
    const int bc   = blockIdx.x;
    const int c    = bc & (CHAN - 1);
    const int w    = threadIdx.x >> 5;
    const int lane = threadIdx.x & 31;
    const float* __restrict__ xb = x  + (size_t)bc * IMG;
    const float* __restrict__ pb = pe + (size_t)c  * IMG;
    float* __restrict__ vs = vsum + (size_t)bc * 4096;
    float* __restrict__ hs = hsum + (size_t)bc * 4096;

    float vacc[2][2][4];
#pragma unroll
    for (int s = 0; s < 2; ++s)
#pragma unroll
        for (int ch = 0; ch < 2; ++ch)
#pragma unroll
            for (int e = 0; e < 4; ++e) vacc[s][ch][e] = 0.f;

    for (int rr = 0; rr < 32; ++rr) {
        const int row = w + 8 * rr;
        const int sel = rr & 1;                    // row&15 == w + 8*sel
        float hp[4] = {0.f, 0.f, 0.f, 0.f};
#pragma unroll
        for (int ch = 0; ch < 2; ++ch) {
            const int col = ch * 128 + lane * 4;
            const float4 xv = *(const float4*)(xb + (size_t)row * NSIDE + col);
            const float4 pv = *(const float4*)(pb + (size_t)row * NSIDE + col);
            float y0 = xv.x + pv.x, y1 = xv.y + pv.y;
            float y2 = xv.z + pv.z, y3 = xv.w + pv.w;
            vacc[sel][ch][0] += y0; vacc[sel][ch][1] += y1;
            vacc[sel][ch][2] += y2; vacc[sel][ch][3] += y3;
            hp[0] += y0; hp[1] += y1; hp[2] += y2; hp[3] += y3;
        }
        // reduce lanes {i, i+4, ..., i+28} -> lane i (i<4); cp = 4*i + e
#pragma unroll
        for (int d = 16; d >= 4; d >>= 1) {
#pragma unroll
            for (int e = 0; e < 4; ++e)
                hp[e] += __shfl_down(hp[e], d, 32);
        }
        if (lane < 4) {
            float4 o; o.x = hp[0]; o.y = hp[1]; o.z = hp[2]; o.w = hp[3];
            *(float4*)(hs + row * 16 + lane * 4) = o;
        }
    }
#pragma unroll
    for (int s = 0; s < 2; ++s) {
        const int rp = w + 8 * s;
#pragma unroll
        for (int ch = 0; ch < 2; ++ch) {
            float4 o;
            o.x = vacc[s][ch][0]; o.y = vacc[s][ch][1];
            o.z = vacc[s][ch][2]; o.w = vacc[s][ch][3];
            *(float4*)(vs + rp * 256 + ch * 128 + lane * 4) = o;
        }
    }
}

// ---------------------------------------------------------------------------
// k2: composite projections via WMMA.  One block per (b,c); waves 0-3 do the
// four d-tiles of vcomp (M=i, N=d, K=256), waves 4-7 the four tiles of hcomp.
// A-fragment layout (f32 16x4): lanes 0-15 carry K={k0,k0+1}, lanes 16-31
// K={k0+2,k0+3}; B mirrors with N = lane&15.
// ---------------------------------------------------------------------------
__global__ __launch_bounds__(256) void k2_proj(const float* __restrict__ vsum,
                                               const float* __restrict__ hsum,
                                               const float* __restrict__ Wv,
                                               const float* __restrict__ bv,
                                               const float* __restrict__ Wh,
                                               const float* __restrict__ bh,
                                               float* __restrict__ vcomp,
                                               float* __restrict__ hcomp) {
    const int bc    = blockIdx.x;
    const int widx  = threadIdx.x >> 5;
    const int lane  = threadIdx.x & 31;
    const bool hpath = widx >= 4;
    const int d0    = (widx & 3) * 16;
    const int lk    = lane & 15;
    const int khalf = (lane >> 4) * 2;           // 0 or 2

    const float* __restrict__ W = hpath ? Wh : Wv;
    const float* __restrict__ A = (hpath ? hsum : vsum) + (size_t)bc * 4096;

    v8f acc = {0.f, 0.f, 0.f, 0.f, 0.f, 0.f, 0.f, 0.f};
#pragma unroll 4
    for (int k0 = 0; k0 < 256; k0 += 4) {
        const int k = k0 + khalf;                // k&15 even -> v2f stays in-row
        v2f a;
        if (hpath) a = *(const v2f*)(A + lk * 256 + k);                       // hsum[j*256+k]
        else       a = *(const v2f*)(A + (k >> 4) * 256 + lk * 16 + (k & 15)); // vsum[rp][i*16+cp]
        const v2f b = *(const v2f*)(W + (d0 + lk) * 256 + k);
        acc = wmma_f32_k4(a, b, acc);
    }
    const int d = d0 + lk;
    const float bias = 16.0f * (hpath ? bh[d] : bv[d]);  // bias summed npatch times
    float* __restrict__ out = (hpath ? hcomp : vcomp) + (size_t)bc * 1024;
    const int mbase = (lane >> 4) * 8;
#pragma unroll
    for (int r = 0; r < 8; ++r)
        out[(mbase + r) * 64 + d] = acc[r] + bias;       // out[i][d], k = i*64+d
}

// ---------------------------------------------------------------------------
// k3: rank projections as two batched GEMMs, M=256 (bc), N=768 (r*256+n),
// K=1024.  1536 wave-tiles of 16x16, 256 WMMA k-steps each, bias fused.
// ---------------------------------------------------------------------------
__global__ __launch_bounds__(256) void k3_gemm(const float* __restrict__ vcomp,
                                               const float* __restrict__ hcomp,
                                               const float* __restrict__ Vw,
                                               const float* __restrict__ Vb,
                                               const float* __restrict__ Hw,
                                               const float* __restrict__ Hb,
                                               float* __restrict__ Vout,
                                               float* __restrict__ Hout) {
    const int wg   = blockIdx.x * 8 + (threadIdx.x >> 5);
    const int lane = threadIdx.x & 31;
    const int g    = wg / 768;                   // 0 = V path, 1 = H path
    const int rem  = wg % 768;
    const int mt   = rem / 48;                   // 16 M-tiles
    const int nt   = rem % 48;                   // 48 N-tiles

    const float* __restrict__ Amat = g ? hcomp : vcomp;
    const float* __restrict__ Bmat = g ? Hw : Vw;       // (RANK,N,KIN) flat: rn*1024+k
    const float* __restrict__ bias = g ? Hb : Vb;       // (RANK,N) flat: rn
    float* __restrict__ Out        = g ? Hout : Vout;

    const int lk    = lane & 15;
    const int khalf = (lane >> 4) * 2;
    const int m  = mt * 16 + lk;                 // bc row
    const int nf = nt * 16 + lk;                 // rn column

    v8f acc = {0.f, 0.f, 0.f, 0.f, 0.f, 0.f, 0.f, 0.f};
#pragma unroll 4
    for (int k0 = 0; k0 < KIN; k0 += 4) {
        const int k = k0 + khalf;
        const v2f a = *(const v2f*)(Amat + (size_t)m  * KIN + k);
        const v2f b = *(const v2f*)(Bmat + (size_t)nf * KIN + k);
        acc = wmma_f32_k4(a, b, acc);
    }
    const float bnf  = bias[nf];
    const int mbase = mt * 16 + (lane >> 4) * 8;
#pragma unroll
    for (int r = 0; r < 8; ++r)
        Out[(size_t)(mbase + r) * RANKN + nf] = acc[r] + bnf;
}

// ---------------------------------------------------------------------------
// k4: out[bc][n][m] = sum_r V[bc][n][r] * H[bc][m][r].  V slice broadcast
// from LDS, H rows in registers, fully coalesced 67 MB output stream.
// ---------------------------------------------------------------------------
__global__ __launch_bounds__(256) void k4_outer(const float* __restrict__ V,
                                                const float* __restrict__ H,
                                                float* __restrict__ out) {
    __shared__ float Vs[RANKN];
    const int bc = blockIdx.x;
    const int t  = threadIdx.x;
    const float* __restrict__ Vb_ = V + (size_t)bc * RANKN;
    const float* __restrict__ Hb_ = H + (size_t)bc * RANKN;
    Vs[t]       = Vb_[t];
    Vs[256 + t] = Vb_[256 + t];
    Vs[512 + t] = Vb_[512 + t];
    const float h0 = Hb_[t], h1 = Hb_[256 + t], h2 = Hb_[512 + t];
    __syncthreads();
    float* __restrict__ ob = out + (size_t)bc * IMG + t;
#pragma unroll 4
    for (int n = 0; n < 256; ++n)
        ob[(size_t)n * 256] = Vs[n] * h0 + Vs[256 + n] * h1 + Vs[512 + n] * h2;
}

// ---------------------------------------------------------------------------
extern "C" void kernel_launch(void* const* d_in, const int* in_sizes, int n_in,
                              void* d_out, int out_size, void* d_ws, size_t ws_size,
                              hipStream_t stream) {
    const float* x   = (const float*)d_in[0];
    const float* pe  = (const float*)d_in[1];
    const float* Wv  = (const float*)d_in[2];
    const float* bv  = (const float*)d_in[3];
    const float* Wh  = (const float*)d_in[4];
    const float* bh  = (const float*)d_in[5];
    const float* Vw  = (const float*)d_in[6];
    const float* Vb  = (const float*)d_in[7];
    const float* Hw  = (const float*)d_in[8];
    const float* Hb  = (const float*)d_in[9];
    float* out = (float*)d_out;
    float* ws  = (float*)d_ws;

    float* vsum  = ws + WS_VSUM;
    float* hsum  = ws + WS_HSUM;
    float* vcomp = ws + WS_VCOMP;
    float* hcomp = ws + WS_HCOMP;
    float* Vws   = ws + WS_V;
    float* Hws   = ws + WS_H;

    k1_reduce<<<NBC, 256, 0, stream>>>(x, pe, vsum, hsum);
    k2_proj <<<NBC, 256, 0, stream>>>(vsum, hsum, Wv, bv, Wh, bh, vcomp, hcomp);
    k3_gemm <<<192, 256, 0, stream>>>(vcomp, hcomp, Vw, Vb, Hw, Hb, Vws, Hws);
    k4_outer<<<NBC, 256, 0, stream>>>(Vws, Hws, out);
}